// Attention_6820408066818
// MI455X (gfx1250) — compile-verified
//
#include <hip/hip_runtime.h>
#include <math.h>

// Problem constants (from reference): B=32, N=100, H=128, BIN=16, E=20000
#define HDIM 128
#define NN 100
#define BB 32
#define BINC 16
#define NPAIR (NN * NN)     // 10000 pairs per batch
#define NEDGE 20000

typedef __attribute__((ext_vector_type(2))) float v2f;
typedef __attribute__((ext_vector_type(8))) float v8f;

// D = A(16x4 f32) * B(4x16 f32) + C(16x16 f32), wave32.
// Probe: _16x16x4_* has 8 args: (neg_a, A, neg_b, B, c_mod, C, reuse_a, reuse_b)
__device__ __forceinline__ v8f wmma_f32(v2f a, v2f b, v8f c) {
  return __builtin_amdgcn_wmma_f32_16x16x4_f32(false, a, false, b, (short)0, c,
                                               false, false);
}

// ---------------------------------------------------------------------------
// Kernel A: y[p, k] = sum_h x[p, h] * W_apair[h, k]   (p = b*N+n, 3200 rows)
// grid = (3200/16) * (128/16) = 1600 waves, 1 wave per 16x16 tile, K=128.
// A-layout (32b 16x4): lane<16 -> {K=0,K=1}, lane>=16 -> {K=2,K=3}, M=lane%16
// B-layout (32b 4x16): mirrored, N=lane%16
// ---------------------------------------------------------------------------
__global__ __launch_bounds__(32) void y_gemm(const float* __restrict__ x,
                                             const float* __restrict__ Wa,
                                             float* __restrict__ y) {
  const int rt = blockIdx.x >> 3;   // 0..199 row tile
  const int ct = blockIdx.x & 7;    // 0..7 col tile
  const int lane = threadIdx.x;
  const int half = lane >> 4, lm = lane & 15;
  const int row = rt * 16 + lm;
  const int c0 = ct * 16;
  v8f acc = {};
  for (int k0 = 0; k0 < HDIM; k0 += 4) {
    const int ka = k0 + 2 * half;
    v2f a, b;
    a.x = x[row * HDIM + ka];
    a.y = x[row * HDIM + ka + 1];
    b.x = Wa[ka * HDIM + c0 + lm];
    b.y = Wa[(ka + 1) * HDIM + c0 + lm];
    acc = wmma_f32(a, b, acc);
  }
#pragma unroll
  for (int v = 0; v < 8; ++v) {
    const int r = rt * 16 + v + 8 * half;  // C/D layout: row = v + 8*(lane/16)
    y[r * HDIM + c0 + lm] = acc[v];
  }
}

// ---------------------------------------------------------------------------
// Kernel B: per 16-pair tile:
//   z = binary[b, p0:p0+16, :16] @ W_binary(16x128)          (WMMA, 8ct x 4kk)
//   att = relu(z + y_i + y_j + b_binary)      (y-sum staged in LDS, 16x128)
//   s[b,p] = sigmoid(att . W_att + b_att)     (shfl_xor reduce over 16 lanes)
// grid = 32 * 625 waves.
// ---------------------------------------------------------------------------
__global__ __launch_bounds__(32) void att_score_kernel(
    const float* __restrict__ binf, const float* __restrict__ y,
    const float* __restrict__ Wb, const float* __restrict__ bb,
    const float* __restrict__ Watt, const float* __restrict__ batt,
    float* __restrict__ s) {
  __shared__ float ysum[16 * HDIM];  // 8 KB of the 320 KB WGP LDS
  const int b = blockIdx.x / 625;
  const int tile = blockIdx.x % 625;
  const int p0 = tile * 16;
  const int lane = threadIdx.x;
  const int half = lane >> 4, lm = lane & 15;

  // Stage ysum[r][c] = y[b, i_r, c] + y[b, j_r, c] + b_binary[c]
  for (int t = 0; t < 64; ++t) {
    const int flat = t * 32 + lane;     // coalesced, 32-wide
    const int r = flat >> 7, c = flat & 127;
    const int p = p0 + r;
    const int i = p / NN, j = p % NN;
    ysum[flat] = y[(b * NN + i) * HDIM + c] + y[(b * NN + j) * HDIM + c] + bb[c];
  }
  __syncthreads();

  // A tiles: 16 pairs x 16 bin-channels, 4 k-steps of 4
  v2f am[4];
  const float* binrow = binf + (size_t)(b * NPAIR + p0 + lm) * BINC;
#pragma unroll
  for (int kk = 0; kk < 4; ++kk) {
    am[kk].x = binrow[kk * 4 + 2 * half];
    am[kk].y = binrow[kk * 4 + 2 * half + 1];
  }

  float wv[8];
#pragma unroll
  for (int ct = 0; ct < 8; ++ct) wv[ct] = Watt[ct * 16 + lm];
  const float ba = batt[0];

  float partial[8];
#pragma unroll
  for (int v = 0; v < 8; ++v) partial[v] = 0.f;

#pragma unroll
  for (int ct = 0; ct < 8; ++ct) {
    v8f acc = {};
#pragma unroll
    for (int kk = 0; kk < 4; ++kk) {
      const int ka = kk * 4 + 2 * half;
      v2f bm;
      bm.x = Wb[ka * HDIM + ct * 16 + lm];
      bm.y = Wb[(ka + 1) * HDIM + ct * 16 + lm];
      acc = wmma_f32(am[kk], bm, acc);
    }
#pragma unroll
    for (int v = 0; v < 8; ++v) {
      const int r = v + 8 * half;                    // C row this lane holds
      float att = acc[v] + ysum[r * HDIM + ct * 16 + lm];
      att = fmaxf(att, 0.f);
      partial[v] += att * wv[ct];
    }
  }
  // Reduce across the 16 lanes of each half (xor masks stay within the half)
#pragma unroll
  for (int v = 0; v < 8; ++v) {
    float p = partial[v];
    p += __shfl_xor(p, 1, 32);
    p += __shfl_xor(p, 2, 32);
    p += __shfl_xor(p, 4, 32);
    p += __shfl_xor(p, 8, 32);
    partial[v] = p;
  }
  if (lm == 0) {
#pragma unroll
    for (int v = 0; v < 8; ++v) {
      const float logit = partial[v] + ba;
      s[b * NPAIR + p0 + v + 8 * half] = 1.f / (1.f + expf(-logit));
    }
  }
}

// ---------------------------------------------------------------------------
// Kernel C: g[b, i, h] = sum_j s[b, i, j] * x[b, j, h]
// Per batch: S(100x100) @ X(100x128). 7 M-tiles (last partial) x 8 N-tiles,
// K=100 in 25 steps of 4. Clamp A-rows, guard stores for rows >= 100.
// ---------------------------------------------------------------------------
__global__ __launch_bounds__(32) void gfeat_gemm(const float* __restrict__ s,
                                                 const float* __restrict__ x,
                                                 float* __restrict__ g) {
  const int b = blockIdx.x / 56;  // 7*8 tiles per batch
  const int t = blockIdx.x % 56;
  const int it = t / 8, ct = t % 8;
  const int lane = threadIdx.x;
  const int half = lane >> 4, lm = lane & 15;
  const int i0 = it * 16;
  const int arow = min(i0 + lm, NN - 1);
  v8f acc = {};
  for (int k0 = 0; k0 < NN; k0 += 4) {  // 100 % 4 == 0
    const int ka = k0 + 2 * half;       // <= 99
    v2f a, bm;
    a.x = s[b * NPAIR + arow * NN + ka];
    a.y = s[b * NPAIR + arow * NN + ka + 1];
    bm.x = x[(b * NN + ka) * HDIM + ct * 16 + lm];
    bm.y = x[(b * NN + ka + 1) * HDIM + ct * 16 + lm];
    acc = wmma_f32(a, bm, acc);
  }
#pragma unroll
  for (int v = 0; v < 8; ++v) {
    const int row = i0 + v + 8 * half;
    if (row < NN) g[(b * NN + row) * HDIM + ct * 16 + lm] = acc[v];
  }
}

// ---------------------------------------------------------------------------
// Kernel D: edge gather. 32 lanes/edge, 4 floats/lane (b128 loads/stores).
// out[0 : E*H]        = local_pair_g[e,h] = x[b,i,h] + x[b,j,h]
// out[E*H : 2*E*H]    = global_pair[e,h]  = g[b,i,h] + g[b,j,h]
// ---------------------------------------------------------------------------
__global__ __launch_bounds__(256) void gather_kernel(
    const int* __restrict__ idx, const float* __restrict__ x,
    const float* __restrict__ g, float* __restrict__ out) {
  const int tid = blockIdx.x * 256 + threadIdx.x;
  const int e = tid >> 5;
  if (e >= NEDGE) return;
  const int h = (tid & 31) * 4;
  const int b = idx[e * 3 + 0], i = idx[e * 3 + 1], j = idx[e * 3 + 2];

  const float4 xi = *(const float4*)(x + ((b * NN + i) * HDIM + h));
  const float4 xj = *(const float4*)(x + ((b * NN + j) * HDIM + h));
  *(float4*)(out + (size_t)e * HDIM + h) =
      make_float4(xi.x + xj.x, xi.y + xj.y, xi.z + xj.z, xi.w + xj.w);

  const float4 gi = *(const float4*)(g + ((b * NN + i) * HDIM + h));
  const float4 gj = *(const float4*)(g + ((b * NN + j) * HDIM + h));
  *(float4*)(out + (size_t)NEDGE * HDIM + (size_t)e * HDIM + h) =
      make_float4(gi.x + gj.x, gi.y + gj.y, gi.z + gj.z, gi.w + gj.w);
}

// ---------------------------------------------------------------------------
// Inputs (setup_inputs order):
//  0 local_feats (32,100,128) f32   4 W_binary (16,128) f32
//  1 binary_feats (32,100,100,16)   5 b_binary (128,)   f32
//  2 sparse_idx (20000,3) int       6 W_att (128,1)     f32
//  3 W_apair (128,128) f32          7 b_att (1,)        f32
// Output: [local_pair_g (E,128) | global_pair (E,128)] f32
// ---------------------------------------------------------------------------
extern "C" void kernel_launch(void* const* d_in, const int* in_sizes, int n_in,
                              void* d_out, int out_size, void* d_ws,
                              size_t ws_size, hipStream_t stream) {
  const float* x    = (const float*)d_in[0];
  const float* binf = (const float*)d_in[1];
  const int*   sidx = (const int*)d_in[2];
  const float* Wa   = (const float*)d_in[3];
  const float* Wb   = (const float*)d_in[4];
  const float* bb   = (const float*)d_in[5];
  const float* Watt = (const float*)d_in[6];
  const float* batt = (const float*)d_in[7];
  float* out = (float*)d_out;

  // Workspace layout (floats): y[3200*128] | s[320000] | g[3200*128]  ~4.6 MB
  float* y = (float*)d_ws;
  float* s = y + 3200 * 128;
  float* g = s + BB * NPAIR;

  hipLaunchKernelGGL(y_gemm, dim3(200 * 8), dim3(32), 0, stream, x, Wa, y);
  hipLaunchKernelGGL(att_score_kernel, dim3(BB * 625), dim3(32), 0, stream,
                     binf, y, Wb, bb, Watt, batt, s);
  hipLaunchKernelGGL(gfeat_gemm, dim3(BB * 56), dim3(32), 0, stream, s, x, g);
  hipLaunchKernelGGL(gather_kernel, dim3((NEDGE * 32 + 255) / 256), dim3(256),
                     0, stream, sidx, x, g, out);
}